// PointProjection_68547678044890
// MI455X (gfx1250) — compile-verified
//
#include <hip/hip_runtime.h>
#include <hip/hip_bf16.h>

typedef __attribute__((ext_vector_type(2))) float v2f;
typedef __attribute__((ext_vector_type(4))) float v4f;
typedef __attribute__((ext_vector_type(8))) float v8f;

// ---------------------------------------------------------------------------
// Stage 1: projection via V_WMMA_F32_16X16X4_F32.
//   ann(3xN) = proMatrix(3x4, zero-padded to 16x4) x p4(4xN), tiled N by 16.
//   One wave handles one 16-point tile. Lanes 0..15 then do the perspective
//   divide + clamp and write hc/wc (the reference's 2nd/3rd outputs).
// A-matrix (16x4 f32, ISA 7.12.2): lanes 0-15 -> {K=0,K=1}, lanes 16-31 -> {K=2,K=3}
// B-matrix (4x16  f32): col = lane&15; lanes 0-15 -> rows {0,1}, lanes 16-31 -> rows {2,3}
// D (16x16 f32, 8 VGPRs): VGPR r: lanes 0-15 = row r, lanes 16-31 = row r+8
// ---------------------------------------------------------------------------
__global__ void proj_wmma_kernel(const float* __restrict__ points,   // (B,N,3)
                                 const float* __restrict__ proM,     // (B,3,4)
                                 float* __restrict__ hc_out,         // (B*N)
                                 float* __restrict__ wc_out,         // (B*N)
                                 int B, int N, int n_tiles) {
  const int lane   = threadIdx.x & 31;
  const int waveId = (blockIdx.x * blockDim.x + threadIdx.x) >> 5;
  if (waveId >= n_tiles) return;                 // uniform per wave: EXEC stays all-1s

  const int tilesPerBatch = N >> 4;
  const int b  = waveId / tilesPerBatch;
  const int n0 = (waveId - b * tilesPerBatch) << 4;

  const int m  = lane & 15;        // row (A) / column (B) within tile
  const int hi = lane >> 4;        // 0: K={0,1}, 1: K={2,3}

  // A = proMatrix padded to 16x4 (rows 3..15 are zero)
  v2f a = {0.0f, 0.0f};
  if (m < 3) {
    const float* Pm = proM + (size_t)b * 12 + (size_t)m * 4 + 2 * hi;
    a.x = Pm[0];
    a.y = Pm[1];
  }

  // B = p4 tile: rows {x, y, z, 1} for point n0+m
  const float* pt = points + ((size_t)b * N + (n0 + m)) * 3;
  v2f bm;
  if (hi == 0) { bm.x = pt[0]; bm.y = pt[1]; }
  else         { bm.x = pt[2]; bm.y = 1.0f;  }

  v8f c = {};
  // D = A x B + 0   (emits v_wmma_f32_16x16x4_f32)
  c = __builtin_amdgcn_wmma_f32_16x16x4_f32(
      /*neg_a=*/false, a, /*neg_b=*/false, bm,
      /*c_mod=*/(short)0, c, /*reuse_a=*/false, /*reuse_b=*/false);

  if (hi == 0) {                                 // lanes 0-15 hold rows 0..7
    const float ann0 = c[0];                     // row 0: w-coord numerator
    const float ann1 = c[1];                     // row 1: h-coord numerator
    const float ann2 = c[2];                     // row 2: depth
    float wcv = ann0 / ann2;
    float hcv = ann1 / ann2;
    // clip(v, 0, 127); NaN propagates like jnp.clip
    wcv = (wcv < 0.0f) ? 0.0f : (wcv > 127.0f ? 127.0f : wcv);
    hcv = (hcv < 0.0f) ? 0.0f : (hcv > 127.0f ? 127.0f : hcv);
    const size_t gp = (size_t)b * N + (n0 + m);
    hc_out[gp] = hcv;
    wc_out[gp] = wcv;
  }
}

// ---------------------------------------------------------------------------
// Stage 2: gather + scale, streaming the 503 MB output.
//   Exact identity used: xi==x1, yi==y1 => w12=w21=w22=0, so
//   out[c] = (x2-x1)*(y2-y1) * feat[c, y1, x1]   (bit-exact vs reference)
//   One thread = 4 consecutive channels of one point -> one NT 128-bit store.
//   Gathers are regular (cached) loads: all feature maps (62.9 MB) fit in L2.
// ---------------------------------------------------------------------------
__global__ void gather_kernel(const float* __restrict__ f0,
                              const float* __restrict__ f1,
                              const float* __restrict__ f2,
                              const float* __restrict__ f3,
                              const float* __restrict__ hc,
                              const float* __restrict__ wc,
                              float* __restrict__ out,     // (B,N,960)
                              int B, int N) {
  const long long total = (long long)B * N * 240;          // 960/4 quads per point
  long long idx = (long long)blockIdx.x * blockDim.x + threadIdx.x;
  if (idx >= total) return;

  const int p  = (int)(idx / 240);                         // global point id = b*N + n
  const int c  = ((int)(idx - (long long)p * 240)) << 2;   // output channel (mult of 4)
  const int b  = p / N;

  const float* __restrict__ f;
  int C, s, coff;
  if (c < 64)       { f = f0; C = 64;  s = 64; coff = 0;   }
  else if (c < 192) { f = f1; C = 128; s = 32; coff = 64;  }
  else if (c < 448) { f = f2; C = 256; s = 16; coff = 192; }
  else              { f = f3; C = 512; s = 8;  coff = 448; }
  const int ch = c - coff;

  const float h = hc[p];
  const float w = wc[p];
  const float sc = (float)s * (1.0f / 128.0f);             // exact power-of-two scale
  const float xs = w * sc;
  const float ys = h * sc;
  const int xi = (int)xs;                                  // == floor == x1 (never clamps)
  const int yi = (int)ys;
  const int x2 = min((int)ceilf(xs), s - 1);
  const int y2 = min((int)ceilf(ys), s - 1);
  const float w11 = (float)((x2 - xi) * (y2 - yi));        // 0 or 1

  const int ss = s * s;
  const size_t base = ((size_t)b * C + ch) * (size_t)ss + (size_t)yi * s + xi;

  v4f v;
  v.x = w11 * f[base];
  v.y = w11 * f[base + ss];
  v.z = w11 * f[base + 2 * (size_t)ss];
  v.w = w11 * f[base + 3 * (size_t)ss];

  // Non-temporal 128-bit store: keep the streamed output out of L2 so the
  // feature maps stay resident.
  __builtin_nontemporal_store(v, reinterpret_cast<v4f*>(out + (size_t)p * 960 + c));
}

extern "C" void kernel_launch(void* const* d_in, const int* in_sizes, int n_in,
                              void* d_out, int out_size, void* d_ws, size_t ws_size,
                              hipStream_t stream) {
  const float* f0     = (const float*)d_in[0];
  const float* f1     = (const float*)d_in[1];
  const float* f2     = (const float*)d_in[2];
  const float* f3     = (const float*)d_in[3];
  const float* points = (const float*)d_in[4];
  const float* proM   = (const float*)d_in[5];

  const int B = in_sizes[5] / 12;            // (B,3,4)
  const int N = in_sizes[4] / (3 * B);       // (B,N,3)

  float* out    = (float*)d_out;
  float* hc_out = out + (size_t)B * N * 960; // outputs concatenated: pixel, hc, wc
  float* wc_out = hc_out + (size_t)B * N;

  // Stage 1: WMMA projection. 1 wave per 16-point tile, 8 waves / 256-thread block.
  const int n_tiles = (B * N) / 16;
  const int blocks1 = (n_tiles + 7) / 8;
  proj_wmma_kernel<<<blocks1, 256, 0, stream>>>(points, proM, hc_out, wc_out,
                                                B, N, n_tiles);

  // Stage 2: gather + NT-streamed output.
  const long long total = (long long)B * N * 240;
  const int blocks2 = (int)((total + 255) / 256);
  gather_kernel<<<blocks2, 256, 0, stream>>>(f0, f1, f2, f3, hc_out, wc_out,
                                             out, B, N);
}